// ModelBLA_61478161875231
// MI455X (gfx1250) — compile-verified
//
#include <hip/hip_runtime.h>
#include <hip/hip_bf16.h>

typedef float v2f __attribute__((ext_vector_type(2)));
typedef float v8f __attribute__((ext_vector_type(8)));

#define NSTEPS 4096
#define RR     256
#define PBLK   64     // time blocks
#define TBLK   64     // steps per block
#define LDA    68     // padded LDS row stride (floats) -> conflict-free ds_load_b64

// ---------------------------------------------------------------------------
// fp32 WMMA 16x16x4:  D = A(16x4) * B(4x16) + C(16x16)
// ---------------------------------------------------------------------------
__device__ __forceinline__ v8f wmma4(v2f a, v2f b, v8f c) {
  return __builtin_amdgcn_wmma_f32_16x16x4_f32(false, a, false, b, (short)0, c,
                                               false, false);
}

__device__ __forceinline__ v2f lds_frag(const float* p, int idx) {
  return *(const v2f*)(p + idx);   // ds_load_b64
}

// Convert state held in 4 D-layout accumulators (64 rows x 16 cols) into 16
// B-operand fragments (K rows 4*kq..4*kq+3).  Pure xor-16 lane permutes.
__device__ __forceinline__ void build_bfrags(const v8f x[4], bool lo, v2f bf[16]) {
#pragma unroll
  for (int m = 0; m < 4; ++m) {
    float s0 = __shfl_xor(x[m][0], 16, 32);
    float s1 = __shfl_xor(x[m][1], 16, 32);
    float s2 = __shfl_xor(x[m][2], 16, 32);
    float s3 = __shfl_xor(x[m][3], 16, 32);
    float s4 = __shfl_xor(x[m][4], 16, 32);
    float s5 = __shfl_xor(x[m][5], 16, 32);
    float s6 = __shfl_xor(x[m][6], 16, 32);
    float s7 = __shfl_xor(x[m][7], 16, 32);
    v2f f;
    f.x = lo ? x[m][0] : s2;  f.y = lo ? x[m][1] : s3;  bf[4*m+0] = f; // rows 0..3
    f.x = lo ? x[m][4] : s6;  f.y = lo ? x[m][5] : s7;  bf[4*m+1] = f; // rows 4..7
    f.x = lo ? s0 : x[m][2];  f.y = lo ? s1 : x[m][3];  bf[4*m+2] = f; // rows 8..11
    f.x = lo ? s4 : x[m][6];  f.y = lo ? s5 : x[m][7];  bf[4*m+3] = f; // rows 12..15
  }
}

// X_new = A*X + B_u*U   (A fragments streamed from LDS, B_u/U in registers)
__device__ __forceinline__ void state_update(v8f x[4], const v2f bf[16],
                                             const float* ldsA, int aoff,
                                             const v2f buf[4][2], const v2f uf[2]) {
  v8f zero = {};
#pragma unroll
  for (int m = 0; m < 4; ++m) {
    v8f acc = wmma4(buf[m][0], uf[0], zero);
    acc = wmma4(buf[m][1], uf[1], acc);
#pragma unroll
    for (int kq = 0; kq < 16; ++kq)
      acc = wmma4(lds_frag(ldsA, m*16*LDA + kq*4 + aoff), bf[kq], acc);
    x[m] = acc;
  }
}

// ---------------------------------------------------------------------------
// K0: A64 = A^64 by repeated squaring (tiny, one workgroup)
// ---------------------------------------------------------------------------
__global__ __launch_bounds__(256) void a_pow_kernel(const float* __restrict__ A,
                                                    float* __restrict__ A64) {
  __shared__ float Ma[64*64];
  __shared__ float Mb[64*64];
  int tid = threadIdx.x;
  for (int i = tid; i < 4096; i += 256) Ma[i] = A[i];
  __syncthreads();
  float* src = Ma;
  float* dst = Mb;
  for (int s = 0; s < 6; ++s) {          // A^2, A^4, ... A^64
    for (int i = tid; i < 4096; i += 256) {
      int r = i >> 6, c = i & 63;
      float acc = 0.f;
#pragma unroll 8
      for (int q = 0; q < 64; ++q) acc += src[r*64 + q] * src[q*64 + c];
      dst[i] = acc;
    }
    __syncthreads();
    float* t = src; src = dst; dst = t;
  }
  for (int i = tid; i < 4096; i += 256) A64[i] = src[i];
}

// ---------------------------------------------------------------------------
// K1 (phase 1): per (time-block j, R-tile t) run T steps from zero state,
// store resulting local input response b_j into workspace.  1024 waves.
// ---------------------------------------------------------------------------
__global__ __launch_bounds__(256) void phase1_kernel(const float* __restrict__ u,
                                                     const float* __restrict__ A,
                                                     const float* __restrict__ B_u,
                                                     float* __restrict__ ws_b) {
  __shared__ float ldsA[64*LDA];
  int tid  = threadIdx.x;
  int lane = tid & 31;
  for (int i = tid; i < 4096; i += 256) ldsA[(i >> 6)*LDA + (i & 63)] = A[i];
  __syncthreads();

  int wg   = blockIdx.x * 8 + (tid >> 5);   // 0..1023
  int j    = wg >> 4;                       // time block
  int tile = wg & 15;                       // R tile
  int lcol = lane & 15, lhi = lane >> 4;
  int aoff = lcol*LDA + (lhi << 1);
  bool lo  = lane < 16;

  v2f buf[4][2];
#pragma unroll
  for (int m = 0; m < 4; ++m)
#pragma unroll
    for (int q = 0; q < 2; ++q)
      buf[m][q] = *(const v2f*)&B_u[(16*m + lcol)*8 + 4*q + 2*lhi];

  v8f x[4] = {};
#pragma unroll 1
  for (int k = 0; k < TBLK; ++k) {
    const float* up = u + (size_t)(j*TBLK + k) * (8*RR) + tile*16 + lcol;
    v2f uf[2];
    uf[0].x = up[(0 + 2*lhi)*RR]; uf[0].y = up[(1 + 2*lhi)*RR];
    uf[1].x = up[(4 + 2*lhi)*RR]; uf[1].y = up[(5 + 2*lhi)*RR];
    if (k < TBLK-1) __builtin_prefetch(up + 8*RR, 0, 1);
    v2f bf[16];
    build_bfrags(x, lo, bf);
    state_update(x, bf, ldsA, aoff, buf, uf);
  }

  float* bp = ws_b + (size_t)j*(64*RR) + tile*16 + lcol + (size_t)(lhi*8)*RR;
#pragma unroll
  for (int m = 0; m < 4; ++m)
#pragma unroll
    for (int r = 0; r < 8; ++r) bp[(16*m + r)*RR] = x[m][r];
}

// ---------------------------------------------------------------------------
// K2 (phase 2): sequential combine over blocks:  S[j+1] = A^64*S[j] + b_j.
// 16 waves (one per R tile), 64 WMMA steps.  Stores S[j] for phase 3.
// ---------------------------------------------------------------------------
__global__ __launch_bounds__(512) void phase2_kernel(const float* __restrict__ x0,
                                                     const float* __restrict__ A64,
                                                     const float* __restrict__ ws_b,
                                                     float* __restrict__ ws_S) {
  __shared__ float ldsA[64*LDA];
  int tid  = threadIdx.x;
  int lane = tid & 31;
  int tile = tid >> 5;                      // 0..15
  for (int i = tid; i < 4096; i += 512) ldsA[(i >> 6)*LDA + (i & 63)] = A64[i];
  __syncthreads();

  int lcol = lane & 15, lhi = lane >> 4;
  int aoff = lcol*LDA + (lhi << 1);
  bool lo  = lane < 16;

  v8f x[4];
  {
    const float* sp = x0 + tile*16 + lcol + (size_t)(lhi*8)*RR;
#pragma unroll
    for (int m = 0; m < 4; ++m)
#pragma unroll
      for (int r = 0; r < 8; ++r) x[m][r] = sp[(16*m + r)*RR];
  }

#pragma unroll 1
  for (int j = 0; j < PBLK; ++j) {
    // emit S[j]
    float* sp = ws_S + (size_t)j*(64*RR) + tile*16 + lcol + (size_t)(lhi*8)*RR;
#pragma unroll
    for (int m = 0; m < 4; ++m)
#pragma unroll
      for (int r = 0; r < 8; ++r) sp[(16*m + r)*RR] = x[m][r];
    // load b_j tile (accumulator layout -> C input of WMMA chain)
    const float* bp = ws_b + (size_t)j*(64*RR) + tile*16 + lcol + (size_t)(lhi*8)*RR;
    v8f bt[4];
#pragma unroll
    for (int m = 0; m < 4; ++m)
#pragma unroll
      for (int r = 0; r < 8; ++r) bt[m][r] = bp[(16*m + r)*RR];

    v2f bf[16];
    build_bfrags(x, lo, bf);
#pragma unroll
    for (int m = 0; m < 4; ++m) {
      v8f acc = bt[m];
#pragma unroll
      for (int kq = 0; kq < 16; ++kq)
        acc = wmma4(lds_frag(ldsA, m*16*LDA + kq*4 + aoff), bf[kq], acc);
      x[m] = acc;
    }
  }
}

// ---------------------------------------------------------------------------
// K3 (phase 3): re-simulate each block from its true initial state S[j],
// emitting Y[k] = C*X + D*U and X[k].  1024 waves.
// ---------------------------------------------------------------------------
__global__ __launch_bounds__(256) void phase3_kernel(const float* __restrict__ u,
                                                     const float* __restrict__ A,
                                                     const float* __restrict__ B_u,
                                                     const float* __restrict__ C_y,
                                                     const float* __restrict__ D_yu,
                                                     const float* __restrict__ ws_S,
                                                     float* __restrict__ outY,
                                                     float* __restrict__ outX) {
  __shared__ float ldsA[64*LDA];
  __shared__ float ldsC[16*LDA];
  int tid  = threadIdx.x;
  int lane = tid & 31;
  for (int i = tid; i < 4096; i += 256) ldsA[(i >> 6)*LDA + (i & 63)] = A[i];
  for (int i = tid; i < 1024; i += 256) {
    int r = i >> 6, c = i & 63;
    ldsC[r*LDA + c] = (r < 8) ? C_y[r*64 + c] : 0.f;   // zero-pad rows 8..15
  }
  __syncthreads();

  int wg   = blockIdx.x * 8 + (tid >> 5);
  int j    = wg >> 4;
  int tile = wg & 15;
  int lcol = lane & 15, lhi = lane >> 4;
  int aoff = lcol*LDA + (lhi << 1);
  bool lo  = lane < 16;

  v2f buf[4][2];
#pragma unroll
  for (int m = 0; m < 4; ++m)
#pragma unroll
    for (int q = 0; q < 2; ++q)
      buf[m][q] = *(const v2f*)&B_u[(16*m + lcol)*8 + 4*q + 2*lhi];

  v2f ddf[2];
#pragma unroll
  for (int q = 0; q < 2; ++q) {
    v2f z; z.x = 0.f; z.y = 0.f;
    ddf[q] = (lcol < 8) ? *(const v2f*)&D_yu[lcol*8 + 4*q + 2*lhi] : z;
  }

  v8f x[4];
  {
    const float* sp = ws_S + (size_t)j*(64*RR) + tile*16 + lcol + (size_t)(lhi*8)*RR;
#pragma unroll
    for (int m = 0; m < 4; ++m)
#pragma unroll
      for (int r = 0; r < 8; ++r) x[m][r] = sp[(16*m + r)*RR];
  }

#pragma unroll 1
  for (int k = 0; k < TBLK; ++k) {
    int kk = j*TBLK + k;
    const float* up = u + (size_t)kk * (8*RR) + tile*16 + lcol;
    v2f uf[2];
    uf[0].x = up[(0 + 2*lhi)*RR]; uf[0].y = up[(1 + 2*lhi)*RR];
    uf[1].x = up[(4 + 2*lhi)*RR]; uf[1].y = up[(5 + 2*lhi)*RR];
    if (k < TBLK-1) __builtin_prefetch(up + 8*RR, 0, 1);

    v2f bf[16];
    build_bfrags(x, lo, bf);

    // store X[kk] (state before update)
    float* xp = outX + (size_t)kk*(64*RR) + tile*16 + lcol + (size_t)(lhi*8)*RR;
#pragma unroll
    for (int m = 0; m < 4; ++m)
#pragma unroll
      for (int r = 0; r < 8; ++r) xp[(16*m + r)*RR] = x[m][r];

    // Y[kk] = C*X + D*U  (C padded to 16 rows; rows 8..15 produce zeros)
    v8f zero = {};
    v8f y = wmma4(ddf[0], uf[0], zero);
    y = wmma4(ddf[1], uf[1], y);
#pragma unroll
    for (int kq = 0; kq < 16; ++kq)
      y = wmma4(lds_frag(ldsC, kq*4 + aoff), bf[kq], y);
    if (lo) {
      float* yp = outY + (size_t)kk*(8*RR) + tile*16 + lane;
#pragma unroll
      for (int r = 0; r < 8; ++r) yp[r*RR] = y[r];
    }

    // X <- A*X + B_u*U
    state_update(x, bf, ldsA, aoff, buf, uf);
  }
}

// ---------------------------------------------------------------------------
extern "C" void kernel_launch(void* const* d_in, const int* in_sizes, int n_in,
                              void* d_out, int out_size, void* d_ws, size_t ws_size,
                              hipStream_t stream) {
  (void)in_sizes; (void)n_in; (void)out_size; (void)ws_size;
  const float* u    = (const float*)d_in[0];
  const float* x0   = (const float*)d_in[1];
  const float* A    = (const float*)d_in[2];
  const float* B_u  = (const float*)d_in[3];
  const float* C_y  = (const float*)d_in[4];
  const float* D_yu = (const float*)d_in[5];

  float* out  = (float*)d_out;
  float* outY = out;                                   // (N, 8, 256)
  float* outX = out + (size_t)NSTEPS * 8 * RR;         // (N, 64, 256)

  float* ws   = (float*)d_ws;
  float* A64  = ws;                                    // 64*64
  float* ws_b = ws + 4096;                             // 64 * 64*256
  float* ws_S = ws_b + (size_t)PBLK * 64 * RR;         // 64 * 64*256

  a_pow_kernel <<<1,   256, 0, stream>>>(A, A64);
  phase1_kernel<<<128, 256, 0, stream>>>(u, A, B_u, ws_b);
  phase2_kernel<<<1,   512, 0, stream>>>(x0, A64, ws_b, ws_S);
  phase3_kernel<<<128, 256, 0, stream>>>(u, A, B_u, C_y, D_yu, ws_S, outY, outX);
}